// MLPDecoder_36644660970266
// MI455X (gfx1250) — compile-verified
//
#include <hip/hip_runtime.h>

typedef float v2f __attribute__((ext_vector_type(2)));
typedef float v8f __attribute__((ext_vector_type(8)));

#define ZD 64   // z feature dim
#define KD 128  // 2*ZD (concat)
#define HD 64   // hidden dim
#define TPW 4   // M-tiles (of 16 edges) per wave -> 64 edges per wave-group

// One wave processes 64 edges (4 M-tiles of 16). Shared B fragments (W1 from
// LDS) are reused across the 4 M-tiles: per K-step = 4 A gathers + 2
// ds_load_2addr_b64 + 16 v_wmma_f32_16x16x4_f32.
__global__ __launch_bounds__(256) void mlp_edge_decoder_kernel(
    const float* __restrict__ z,          // [N_NODES, 64]
    const long long* __restrict__ edge,   // [2, E] int64
    const float* __restrict__ W1,         // [128, 64] row-major (in, out)
    const float* __restrict__ b1,         // [64]
    const float* __restrict__ W2,         // [64]
    const float* __restrict__ b2,         // [1]
    float* __restrict__ out,              // [E]
    int E)
{
    // W1 staged in LDS, K-quad interleaved so each lane's B fragment is one
    // conflict-free b64: element (k,n) -> w1s[((k>>2)*64 + n)*4 + (k&3)]
    __shared__ float w1s[KD * HD];

    const int tid = threadIdx.x;
    for (int i = tid; i < KD * HD; i += 256) {
        const int k = i >> 6;
        const int n = i & 63;
        w1s[(((k >> 2) * HD + n) << 2) + (k & 3)] = W1[i];
    }
    __syncthreads();

    const int lane = tid & 31;
    const int wave = tid >> 5;
    const int m    = lane & 15;   // row within M-tile / N-col within N-tile
    const int hl   = lane >> 4;   // lane half: K{0,1} vs K{2,3} of each quad

    // Per-lane epilogue constants (C/D layout: lane holds N = m + 16*nt)
    float bias[4], w2v[4];
#pragma unroll
    for (int nt = 0; nt < 4; ++nt) {
        bias[nt] = b1[nt * 16 + m];
        w2v[nt]  = W2[nt * 16 + m];
    }
    const float b2v = b2[0];

    const long long* __restrict__ esrc = edge;
    const long long* __restrict__ edst = edge + E;

    const int nGroups    = E >> 6;                         // 64 edges/group, E%64==0
    const int waveGlobal = blockIdx.x * (blockDim.x >> 5) + wave;
    const int waveStride = gridDim.x * (blockDim.x >> 5);

    for (int g = waveGlobal; g < nGroups; g += waveStride) {
        const int e0 = g << 6;

        // Gather row pointers for all 4 M-tiles (both lane halves need row m)
        const float* zsrc[TPW];
        const float* zdst[TPW];
#pragma unroll
        for (int t = 0; t < TPW; ++t) {
            const long long si = esrc[e0 + t * 16 + m];
            const long long di = edst[e0 + t * 16 + m];
            zsrc[t] = z + si * ZD;
            zdst[t] = z + di * ZD;
        }

        // Accumulators pre-loaded with bias
        v8f acc[TPW][4];
#pragma unroll
        for (int t = 0; t < TPW; ++t)
#pragma unroll
            for (int nt = 0; nt < 4; ++nt)
#pragma unroll
                for (int r = 0; r < 8; ++r)
                    acc[t][nt][r] = bias[nt];

        // ---- Phase 1: K = 0..63 (zi features) ----
#pragma unroll 2
        for (int kk = 0; kk < 16; ++kk) {
            v2f a[TPW];
#pragma unroll
            for (int t = 0; t < TPW; ++t)
                a[t] = *(const v2f*)(zsrc[t] + 4 * kk + 2 * hl);
            v2f b[4];
#pragma unroll
            for (int nt = 0; nt < 4; ++nt)
                b[nt] = *(const v2f*)(&w1s[((kk * HD + nt * 16 + m) << 2) + 2 * hl]);
#pragma unroll
            for (int t = 0; t < TPW; ++t)
#pragma unroll
                for (int nt = 0; nt < 4; ++nt)
                    acc[t][nt] = __builtin_amdgcn_wmma_f32_16x16x4_f32(
                        false, a[t], false, b[nt], (short)0, acc[t][nt], false, false);
        }
        // ---- Phase 2: K = 64..127 (zj features) ----
#pragma unroll 2
        for (int kk = 0; kk < 16; ++kk) {
            v2f a[TPW];
#pragma unroll
            for (int t = 0; t < TPW; ++t)
                a[t] = *(const v2f*)(zdst[t] + 4 * kk + 2 * hl);
            v2f b[4];
#pragma unroll
            for (int nt = 0; nt < 4; ++nt)
                b[nt] = *(const v2f*)(&w1s[(((kk + 16) * HD + nt * 16 + m) << 2) + 2 * hl]);
#pragma unroll
            for (int t = 0; t < TPW; ++t)
#pragma unroll
                for (int nt = 0; nt < 4; ++nt)
                    acc[t][nt] = __builtin_amdgcn_wmma_f32_16x16x4_f32(
                        false, a[t], false, b[nt], (short)0, acc[t][nt], false, false);
        }

        // ---- ReLU + h @ W2 + b2, per M-tile ----
        // D layout: reg r, lanes 0-15 -> M=r, lanes 16-31 -> M=r+8; N = m + 16*nt
#pragma unroll
        for (int t = 0; t < TPW; ++t) {
#pragma unroll
            for (int r = 0; r < 8; ++r) {
                float p = 0.f;
#pragma unroll
                for (int nt = 0; nt < 4; ++nt)
                    p += fmaxf(acc[t][nt][r], 0.f) * w2v[nt];
                // butterfly reduce across the 16 N-lanes of each half
                p += __shfl_xor(p, 1);
                p += __shfl_xor(p, 2);
                p += __shfl_xor(p, 4);
                p += __shfl_xor(p, 8);
                if (m == 0)
                    out[e0 + t * 16 + hl * 8 + r] = p + b2v;
            }
        }
    }
}

extern "C" void kernel_launch(void* const* d_in, const int* in_sizes, int n_in,
                              void* d_out, int out_size, void* d_ws, size_t ws_size,
                              hipStream_t stream) {
    const float*     z    = (const float*)d_in[0];
    const long long* edge = (const long long*)d_in[1];
    const float*     W1   = (const float*)d_in[2];
    const float*     b1   = (const float*)d_in[3];
    const float*     W2   = (const float*)d_in[4];
    const float*     b2   = (const float*)d_in[5];
    float*           out  = (float*)d_out;

    const int E = out_size;           // == in_sizes[1] / 2
    const int blocks = 1024;          // 8192 waves; ~1.9 groups (64 edges) each

    mlp_edge_decoder_kernel<<<blocks, 256, 0, stream>>>(z, edge, W1, b1, W2, b2, out, E);
}